// BuzzLoss_4638564679732
// MI455X (gfx1250) — compile-verified
//
#include <hip/hip_runtime.h>
#include <cstdint>

// BuzzLoss: per-row sequential scan over T=1024, reduced over B rows to a scalar.
// Memory-bound: 64MB / 23.3 TB/s ~= 2.75us. CDNA5 path: async global->LDS
// streaming (ASYNCcnt) + wave32 shuffle prefix-product scan.

#define T_LEN 1024
#define BLOCK_THREADS 128
#define WAVES_PER_BLOCK 4
#define GRID_BLOCKS 512
#define TILES (T_LEN / 32)

__device__ __forceinline__ uint32_t lds_byte_offset(const void* p) {
    // Generic pointer to LDS: low 32 bits are the LDS byte offset (aperture in high bits).
    return (uint32_t)(uintptr_t)p;
}

// Stream one row (conf 4KB + acc 4KB) into LDS: 16 x global_load_async_to_lds_b128,
// each moving 32 lanes x 16B = 512B fully coalesced, tracked on ASYNCcnt.
__device__ __forceinline__ void issue_row_async(const float* g_conf_row,
                                                const float* g_acc_row,
                                                uint32_t lds_conf, uint32_t lds_acc,
                                                int lane) {
    uint64_t gc = (uint64_t)(uintptr_t)g_conf_row + (uint64_t)(lane * 16);
    uint64_t ga = (uint64_t)(uintptr_t)g_acc_row  + (uint64_t)(lane * 16);
    uint32_t lc = lds_conf + (uint32_t)(lane * 16);
    uint32_t la = lds_acc  + (uint32_t)(lane * 16);
    asm volatile(
        "global_load_async_to_lds_b128 %0, %2, off offset:0\n\t"
        "global_load_async_to_lds_b128 %0, %2, off offset:512\n\t"
        "global_load_async_to_lds_b128 %0, %2, off offset:1024\n\t"
        "global_load_async_to_lds_b128 %0, %2, off offset:1536\n\t"
        "global_load_async_to_lds_b128 %0, %2, off offset:2048\n\t"
        "global_load_async_to_lds_b128 %0, %2, off offset:2560\n\t"
        "global_load_async_to_lds_b128 %0, %2, off offset:3072\n\t"
        "global_load_async_to_lds_b128 %0, %2, off offset:3584\n\t"
        "global_load_async_to_lds_b128 %1, %3, off offset:0\n\t"
        "global_load_async_to_lds_b128 %1, %3, off offset:512\n\t"
        "global_load_async_to_lds_b128 %1, %3, off offset:1024\n\t"
        "global_load_async_to_lds_b128 %1, %3, off offset:1536\n\t"
        "global_load_async_to_lds_b128 %1, %3, off offset:2048\n\t"
        "global_load_async_to_lds_b128 %1, %3, off offset:2560\n\t"
        "global_load_async_to_lds_b128 %1, %3, off offset:3072\n\t"
        "global_load_async_to_lds_b128 %1, %3, off offset:3584\n\t"
        :
        : "v"(lc), "v"(la), "v"(gc), "v"(ga)
        : "memory");
}

__global__ void __launch_bounds__(BLOCK_THREADS)
buzz_scan_kernel(const float* __restrict__ conf, const float* __restrict__ acc,
                 float* __restrict__ scores, int B) {
    // [wave][stage][conf/acc][T]
    __shared__ __align__(16) float lds[WAVES_PER_BLOCK][2][2][T_LEN];

    const int lane = threadIdx.x & 31;
    const int wave = threadIdx.x >> 5;
    const int waves_total = GRID_BLOCKS * WAVES_PER_BLOCK;
    const int r0 = blockIdx.x * WAVES_PER_BLOCK + wave;

    uint32_t lb[2][2];
    lb[0][0] = lds_byte_offset(&lds[wave][0][0][0]);
    lb[0][1] = lds_byte_offset(&lds[wave][0][1][0]);
    lb[1][0] = lds_byte_offset(&lds[wave][1][0][0]);
    lb[1][1] = lds_byte_offset(&lds[wave][1][1][0]);

    int stage = 0;
    if (r0 < B) {
        issue_row_async(conf + (size_t)r0 * T_LEN, acc + (size_t)r0 * T_LEN,
                        lb[0][0], lb[0][1], lane);
    }

    for (int r = r0; r < B; r += waves_total) {
        const int rn = r + waves_total;
        if (rn < B) {
            // Prefetch next row into the other stage, then wait until only the
            // prefetch's 16 ops remain outstanding (async loads retire in order).
            issue_row_async(conf + (size_t)rn * T_LEN, acc + (size_t)rn * T_LEN,
                            lb[stage ^ 1][0], lb[stage ^ 1][1], lane);
            asm volatile("s_wait_asynccnt 16" ::: "memory");
        } else {
            asm volatile("s_wait_asynccnt 0" ::: "memory");
        }

        const float* cb = &lds[wave][stage][0][0];
        const float* ab = &lds[wave][stage][1][0];

        float p = 1.0f;   // running product of (1-conf) over completed tiles (wave-uniform)
        float s = 0.0f;   // per-lane partial of sum(conf * excl_cumprod * acc)

        for (int t = 0; t < TILES; ++t) {
            float c  = cb[t * 32 + lane];   // coalesced, bank-conflict-free
            float a  = ab[t * 32 + lane];
            float nb = 1.0f - c;

            // Inclusive prefix product of nb across the wave (log2(32)=5 steps).
            float pp = nb;
            #pragma unroll
            for (int d = 1; d < 32; d <<= 1) {
                float o = __shfl_up(pp, (unsigned)d, 32);
                if (lane >= d) pp *= o;
            }
            // Exclusive within-tile product for this lane.
            float excl = __shfl_up(pp, 1u, 32);
            if (lane == 0) excl = 1.0f;

            s = fmaf(p, c * excl * a, s);       // buzz_prob * acc contribution
            p *= __shfl(pp, 31, 32);            // fold in the whole tile's product
        }

        const float aLast = ab[T_LEN - 1];      // LDS broadcast read

        // Reduce per-lane partials across the wave.
        #pragma unroll
        for (int d = 16; d > 0; d >>= 1) s += __shfl_xor(s, d, 32);

        if (lane == 0) {
            // correction = 1 - sum(buzz) = full product p (telescoping)
            scores[r] = s + p * aLast;
        }
        stage ^= 1;
    }
}

// Deterministic fixed-order reduction of B row scores -> -mean.
__global__ void __launch_bounds__(256)
buzz_reduce_kernel(const float* __restrict__ scores, float* __restrict__ out, int B) {
    __shared__ float sm[256];
    const int t = threadIdx.x;
    float s = 0.0f;
    for (int i = t; i < B; i += 256) s += scores[i];
    sm[t] = s;
    __syncthreads();
    for (int w = 128; w > 0; w >>= 1) {
        if (t < w) sm[t] += sm[t + w];
        __syncthreads();
    }
    if (t == 0) out[0] = -sm[0] / (float)B;
}

extern "C" void kernel_launch(void* const* d_in, const int* in_sizes, int n_in,
                              void* d_out, int out_size, void* d_ws, size_t ws_size,
                              hipStream_t stream) {
    const float* conf = (const float*)d_in[0];
    const float* acc  = (const float*)d_in[1];
    const int B = in_sizes[0] / T_LEN;   // reference: B=8192, T=1024
    float* scores = (float*)d_ws;        // B floats of scratch

    buzz_scan_kernel<<<GRID_BLOCKS, BLOCK_THREADS, 0, stream>>>(conf, acc, scores, B);
    buzz_reduce_kernel<<<1, 256, 0, stream>>>(scores, (float*)d_out, B);
}